// AdaptiveGraphConv_71554155151644
// MI455X (gfx1250) — compile-verified
//
#include <hip/hip_runtime.h>
#include <hip/hip_bf16.h>

typedef __attribute__((ext_vector_type(2))) float v2f;
typedef __attribute__((ext_vector_type(8))) float v8f;
typedef unsigned int u32x4 __attribute__((ext_vector_type(4)));
typedef int i32x8 __attribute__((ext_vector_type(8)));
typedef int i32x4 __attribute__((ext_vector_type(4)));

// Problem constants (from reference setup_inputs)
#define NN 128
#define CC 64
#define TT 300
#define VV 25
#define KK 3
#define OO 64
#define EE 8

#define TILE_T 8          // timesteps per block
#define T_TILES 38        // ceil(300/8)
#define VPAD 28           // K-steps: 7 x 4 (adjacency rows 25..27 zeroed)
#define WPAD 32           // w pad (2 N-tiles of 16)

// LDS partition (floats).  Ys2 (hot, largest offsets) at base 0 so its
// biggest byte offset (~65.5KB) fits the 16-bit DS immediate.
#define YS_SZ (512 * WPAD)        // 16384 : Ys2[c/2][col(t,w)][2] pair-interleaved K
#define AS_SZ (5 * VPAD * WPAD)   // 4480  : As2[k][v/2][w][2]     pair-interleaved K
#define XS_SZ (512 * VV + 8)      // 12808 : Xs[row=(c*8+tl)*25 + v] (TDM layout) + slack

// ---------------------------------------------------------------------------
// Prep 1: degree-normalized learned adjacency  D^{-1/2} |A| D^{-1/2}  (25x25)
// ---------------------------------------------------------------------------
__global__ void learned_adj_kernel(const float* __restrict__ A_learned,
                                   float* __restrict__ A_l) {
    __shared__ float d[VV];
    int tid = threadIdx.x;
    if (tid < VV) {
        float s = 0.f;
        for (int w = 0; w < VV; ++w) s += fabsf(A_learned[tid * VV + w]);
        s = fmaxf(s, 1e-6f);
        d[tid] = rsqrtf(s);     // clip(sum)^-0.5
    }
    __syncthreads();
    for (int i = tid; i < VV * VV; i += blockDim.x) {
        int v = i / VV, w = i % VV;
        A_l[i] = d[v] * fabsf(A_learned[i]) * d[w];
    }
}

// ---------------------------------------------------------------------------
// Prep 2: per-sample dynamic adjacency (temporal pool -> embed -> cosine ->
// softmax).  One block per sample n.
// ---------------------------------------------------------------------------
__global__ void dyn_adj_kernel(const float* __restrict__ x,
                               const float* __restrict__ W_node,
                               float* __restrict__ A_dyn) {
    const int n = blockIdx.x;
    __shared__ float pool[CC * VV];     // (c,v) temporal means
    __shared__ float part[10 * VV];
    __shared__ float emb[VV * EE];
    __shared__ float nrm[VV];

    const int tid = threadIdx.x;
    const int grp = tid / VV;           // 0..9 active (250 threads)
    const int v   = tid % VV;

    for (int c = 0; c < CC; ++c) {
        float s = 0.f;
        if (grp < 10) {
            const float* xp = x + ((size_t)(n * CC + c) * TT) * VV + v;
            for (int t = grp; t < TT; t += 10) s += xp[t * VV];
            part[grp * VV + v] = s;
        }
        __syncthreads();
        if (grp == 0) {
            float a = 0.f;
            for (int g = 0; g < 10; ++g) a += part[g * VV + v];
            pool[c * VV + v] = a * (1.0f / (float)TT);
        }
        __syncthreads();
    }

    // emb[v][e] = sum_c pool[c][v] * W_node[e][c]
    if (tid < VV * EE) {
        int vv = tid >> 3, e = tid & 7;
        float s = 0.f;
        for (int c = 0; c < CC; ++c) s += pool[c * VV + vv] * W_node[e * CC + c];
        emb[vv * EE + e] = s;
    }
    __syncthreads();
    if (tid < VV) {
        float s = 0.f;
        for (int e = 0; e < EE; ++e) { float t = emb[tid * EE + e]; s += t * t; }
        nrm[tid] = fmaxf(sqrtf(s), 1e-12f);
    }
    __syncthreads();
    // cosine scores + row softmax (25 rows, one thread each)
    if (tid < VV) {
        float sc[VV];
        float mx = -1e30f;
        for (int w = 0; w < VV; ++w) {
            float s = 0.f;
            for (int e = 0; e < EE; ++e) s += emb[tid * EE + e] * emb[w * EE + e];
            s /= (nrm[tid] * nrm[w]);
            sc[w] = s;
            mx = fmaxf(mx, s);
        }
        float sum = 0.f;
        for (int w = 0; w < VV; ++w) { float e = __expf(sc[w] - mx); sc[w] = e; sum += e; }
        float inv = 1.0f / sum;
        for (int w = 0; w < VV; ++w) A_dyn[(size_t)n * (VV * VV) + tid * VV + w] = sc[w] * inv;
    }
}

// ---------------------------------------------------------------------------
// Main fused kernel: one block = (sample n, 8-timestep tile), 8 waves.
// x tile staged via the Tensor Data Mover (one 2D D#: 64 rows x 200 floats,
// row stride 7500).  For each of 5 branches k:
//   GEMM1 (WMMA f32 16x16x4): Y[(c,t),w] = X[(c,t),v] * A_k[v,w]   (LDS->LDS)
//   GEMM2 (WMMA f32 16x16x4): acc[o,(t,w)] += W_k[o,c] * Y[c,(t,w)]
// Accumulators persist across k; output written once, w-major coalesced.
// ---------------------------------------------------------------------------
__global__ void __launch_bounds__(256)
adaptive_gconv_kernel(const float* __restrict__ x,
                      const float* __restrict__ A_physical,
                      const float* __restrict__ W_subsets,
                      const float* __restrict__ A_l,
                      const float* __restrict__ A_dyn,
                      float* __restrict__ out) {
    extern __shared__ float smem[];
    float* Ys2 = smem;                  // [32][256][2]   (c-pair, col=t*32+w, p)
    float* As2 = smem + YS_SZ;          // [5][14][32][2] (k, v-pair, w, p)
    float* Xs  = smem + YS_SZ + AS_SZ;  // [512*25 + 8]   (row=(c*8+tl), v) stride 25

    const int tile = blockIdx.x;        // 0..37
    const int n    = blockIdx.y;        // 0..127
    const int t0   = tile * TILE_T;

    const int tid  = threadIdx.x;
    const int wave = tid >> 5;
    const int lane = tid & 31;
    const int l15  = lane & 15;
    const int lh   = lane >> 4;         // 0/1: fragment half selector

    // ---- kick off TDM load of the x tile (wave 0 issues the descriptor) ----
#if __has_builtin(__builtin_amdgcn_tensor_load_to_lds)
    if (wave == 0) {
        const float* gsrc = x + (size_t)n * (CC * TT * VV) + (size_t)t0 * VV;
        unsigned long long ga = (unsigned long long)(uintptr_t)gsrc;
        unsigned xs_lds = (unsigned)(uintptr_t)(void*)Xs;      // low 32b = LDS addr
        unsigned rem = (unsigned)(TT * VV - t0 * VV);          // OOB zero-fill tail

        u32x4 g0;
        g0[0] = 1u;                                            // count=1 (user D#)
        g0[1] = xs_lds;                                        // lds_addr
        g0[2] = (unsigned)(ga & 0xFFFFFFFFu);                  // global_addr[31:0]
        g0[3] = (unsigned)((ga >> 32) & 0x1FFFFFFu) | (2u << 30); // ga[56:32]|type=2

        i32x8 g1;
        g1[0] = (int)(2u << 16);                 // wg_mask=0, data_size=4B
        g1[1] = (int)(rem << 16);                // tensor_dim0[15:0]  @bits63:48
        g1[2] = (int)((rem >> 16) | (64u << 16));// tensor_dim0[31:16] | tensor_dim1=64
        g1[3] = (int)(200u << 16);               // tile_dim0=200 (8t x 25v)
        g1[4] = (int)(64u);                      // tile_dim1=64, tile_dim2=0
        g1[5] = (int)(7500u);                    // tensor_dim0_stride = T*V
        g1[6] = 0;
        g1[7] = 0;
        i32x4 gz4 = {0, 0, 0, 0};
        i32x8 gz8 = {0, 0, 0, 0, 0, 0, 0, 0};
        __builtin_amdgcn_tensor_load_to_lds(g0, g1, gz4, gz4, gz8, 0);
    }
#else
    for (int i = tid; i < 512 * VV; i += 256) {
        int row = i / VV;
        int v   = i % VV;
        int c   = row >> 3;
        int t   = t0 + (row & 7);
        Xs[i] = (t < TT) ? x[((size_t)(n * CC + c) * TT + t) * VV + v] : 0.f;
    }
#endif

    // ---- stage adjacency stack, pair-interleaved + zero padded -------------
    // As2 idx = k*896 + v2*64 + w*2 + p ; v = 2*v2+p ; rows v>=25, cols w>=25 zero
#pragma unroll
    for (int k = 0; k < 5; ++k) {
        const float* src = (k < KK) ? (A_physical + k * (VV * VV))
                         : (k == KK ? A_l : A_dyn + (size_t)n * (VV * VV));
        for (int i = tid; i < VPAD * WPAD; i += 256) {
            int v = ((i >> 6) << 1) | (i & 1);
            int w = (i & 63) >> 1;
            As2[k * (VPAD * WPAD) + i] = (v < VV && w < VV) ? src[v * VV + w] : 0.f;
        }
    }

    // prefetch next t-tile of x into the cache hierarchy
    if (t0 + TILE_T < TT) {
        for (int i = tid; i < 512; i += 256) {
            int c = i >> 3, tl = i & 7;
            __builtin_prefetch(&x[((size_t)(n * CC + c) * TT + (t0 + TILE_T + tl)) * VV], 0, 1);
        }
    }

#if __has_builtin(__builtin_amdgcn_tensor_load_to_lds)
    if (wave == 0) __builtin_amdgcn_s_wait_tensorcnt(0);
#endif
    __syncthreads();

    // persistent GEMM2 accumulators: wave owns o-rows [(wave>>1)*16, +16),
    // col tiles ((wave&1)*8 + j)*16 for j in 0..7  (cols = t*32 + w)
    v8f acc[8];
#pragma unroll
    for (int j = 0; j < 8; ++j) acc[j] = (v8f)(0.f);

    const int oRow = (wave >> 1) * 16 + l15;     // A-row (W) for GEMM2
    const int colB = (wave & 1) * 8;             // first col-tile index

    for (int k = 0; k < 5; ++k) {
        const float* Ak2 = As2 + k * (VPAD * WPAD);        // k*896
        const float* Wk  = W_subsets + (size_t)k * OO * CC;

        // ---- GEMM1: Y = X * A_k ; wave handles 4 row-tiles x 2 w-tiles -----
        // Preload all adjacency B-fragments (7 ksteps x 2 w-tiles), b64 each.
        v2f bf[7][2];
#pragma unroll
        for (int kk = 0; kk < 7; ++kk)
#pragma unroll
            for (int wt = 0; wt < 2; ++wt)
                bf[kk][wt] = *(const v2f*)&Ak2[(kk * 2 + lh) * (WPAD * 2) +
                                               (wt * 16 + l15) * 2];

#pragma unroll
        for (int rt = 0; rt < 4; ++rt) {
            const int row0 = (wave * 4 + rt) * 16;
            v2f af[7];
            const float* xr = &Xs[(row0 + l15) * VV + lh * 2];
#pragma unroll
            for (int kk = 0; kk < 7; ++kk) {    // 4B-aligned pair loads (2addr)
                af[kk][0] = xr[kk * 4];
                af[kk][1] = xr[kk * 4 + 1];
            }

#pragma unroll
            for (int wt = 0; wt < 2; ++wt) {
                v8f y = (v8f)(0.f);
#pragma unroll
                for (int kk = 0; kk < 7; ++kk)
                    y = __builtin_amdgcn_wmma_f32_16x16x4_f32(
                            false, af[kk], false, bf[kk][wt], (short)0, y,
                            false, false);
                // scatter D to Ys2: element (row=rowd, w) -> [c/2][tl*32+w][c&1]
#pragma unroll
                for (int r = 0; r < 8; ++r) {
                    int rowd = row0 + r + lh * 8;
                    int c    = rowd >> 3;
                    int col  = (rowd & 7) * 32 + wt * 16 + l15;
                    Ys2[(c >> 1) * 512 + col * 2 + (c & 1)] = y[r];
                }
            }
        }
        __syncthreads();   // Ys2 complete for this branch

        // ---- GEMM2: acc += W_k (64x64) * Y (64 x 256) ----------------------
        const float* wrow = &Wk[oRow * CC + lh * 2];
#pragma unroll
        for (int c0 = 0; c0 < CC; c0 += 4) {
            v2f aw = *(const v2f*)&wrow[c0];               // global b64
            const float* yb = &Ys2[(c0 / 2 + lh) * 512 + l15 * 2];
#pragma unroll
            for (int j = 0; j < 8; ++j) {
                const int col0 = (colB + j) * 16;
                v2f b = *(const v2f*)&yb[col0 * 2];        // ds_load_b64
                acc[j] = __builtin_amdgcn_wmma_f32_16x16x4_f32(
                             false, aw, false, b, (short)0, acc[j],
                             false, false);
            }
        }
        __syncthreads();   // done reading Ys2 before next branch overwrites
    }

    // ---- store: acc tiles -> out[n][o][t][w], drop w>=25 / t>=300 ----------
#pragma unroll
    for (int j = 0; j < 8; ++j) {
        const int col0 = (colB + j) * 16;
#pragma unroll
        for (int r = 0; r < 8; ++r) {
            int o   = (wave >> 1) * 16 + r + lh * 8;
            int col = col0 + l15;
            int tl  = col >> 5;
            int w   = col & 31;
            int t   = t0 + tl;
            if (w < VV && t < TT)
                out[((size_t)(n * OO + o) * TT + t) * VV + w] = acc[j][r];
        }
    }
}

// ---------------------------------------------------------------------------
extern "C" void kernel_launch(void* const* d_in, const int* in_sizes, int n_in,
                              void* d_out, int out_size, void* d_ws, size_t ws_size,
                              hipStream_t stream) {
    const float* x          = (const float*)d_in[0];
    const float* A_physical = (const float*)d_in[1];
    const float* A_learned  = (const float*)d_in[2];
    const float* W_subsets  = (const float*)d_in[3];
    const float* W_node     = (const float*)d_in[4];
    float* out = (float*)d_out;

    float* A_l   = (float*)d_ws;        // 625 floats (pad to 640)
    float* A_dyn = A_l + 640;           // 128*625 floats

    learned_adj_kernel<<<1, 256, 0, stream>>>(A_learned, A_l);
    dyn_adj_kernel<<<NN, 256, 0, stream>>>(x, W_node, A_dyn);

    size_t smem = (size_t)(YS_SZ + AS_SZ + XS_SZ) * sizeof(float); // ~134.7 KB
    dim3 grid(T_TILES, NN);
    adaptive_gconv_kernel<<<grid, 256, smem, stream>>>(
        x, A_physical, W_subsets, A_l, A_dyn, out);
}